// QuantumEvolutionNetwork_19207093747837
// MI455X (gfx1250) — compile-verified
//
#include <hip/hip_runtime.h>
#include <hip/hip_bf16.h>
#include <stdint.h>

#define DIM 2048
#define BATCH 4096
#define NS_ITERS 14   // Newton-Schulz polar iterations per layer (GEMM-only unitarization)

typedef __attribute__((ext_vector_type(2))) float v2f;
typedef __attribute__((ext_vector_type(8))) float v8f;
typedef __attribute__((ext_vector_type(4))) int   v4i;

#define BM 128
#define BN 64
#define BK 16
#define APITCH  (BK + 4)     // A as [m][k] (normal): 128 x 20, 80B rows (16B aligned)
#define TAPITCH (BM + 4)     // A as [k][m] (conjA):  16 x 132, 528B rows (16B aligned)
#define AMAX    (BM * APITCH)  // 2560 floats >= 16*132 = 2112
#define BPITCH  (BN + 4)     // 68 floats, 272B rows (16B aligned)
#define BSZ     (BK * BPITCH)

#define GLOBAL_AS __attribute__((address_space(1)))
#define LDS_AS    __attribute__((address_space(3)))

// Async global->LDS B128 copy (per-lane LDS destination), tracked by ASYNCcnt.
__device__ __forceinline__ void async_b128(void* lds, const void* g) {
#if __has_builtin(__builtin_amdgcn_global_load_async_to_lds_b128)
  __builtin_amdgcn_global_load_async_to_lds_b128(
      (GLOBAL_AS v4i*)g, (LDS_AS v4i*)lds, /*offset=*/0, /*cpol=*/0);
#else
  asm volatile("global_load_async_to_lds_b128 %0, %1, off"
               :: "v"((uint32_t)(uintptr_t)lds), "v"(g) : "memory");
#endif
}

__device__ __forceinline__ void wait_async0() {
#if __has_builtin(__builtin_amdgcn_s_wait_asynccnt)
  __builtin_amdgcn_s_wait_asynccnt(0);
#else
  asm volatile("s_wait_asynccnt 0x0" ::: "memory");
#endif
}

__device__ __forceinline__ v8f wmma_f32x4(v2f a, v2f b, v8f c) {
  // V_WMMA_F32_16X16X4_F32 : D = A(16x4) * B(4x16) + C(16x16), wave32
  return __builtin_amdgcn_wmma_f32_16x16x4_f32(
      /*neg_a=*/false, a, /*neg_b=*/false, b,
      /*c_mod=*/(short)0, c, /*reuse_a=*/false, /*reuse_b=*/false);
}

// Complex GEMM: D = alpha * (op(A) * B) + diag_add * I   (op = N or conjugate-transpose)
// A is DIM x DIM (row-major). B,D are DIM x N (row-major).
// Karatsuba complex MAC: M1=Ar*Br, M2=Ai*Bi, M3=(Ar+s*Ai)(Br+Bi), s=+-1 for op(A);
//   re = M1 - s*M2 ; im = M3 - M1 - s*M2   -> 3 WMMAs per k-step instead of 4.
__global__ __launch_bounds__(256)
void cgemm_wmma_kernel(const float* __restrict__ Ar, const float* __restrict__ Ai, int conjA,
                       const float* __restrict__ Br, const float* __restrict__ Bi,
                       float* __restrict__ Dr, float* __restrict__ Di,
                       int N, float alpha, const float* alpha_ptr, float diag_add)
{
  __shared__ float lsA[2][2][AMAX];   // [buf][re/im]
  __shared__ float lsB[2][2][BSZ];

  const int tid  = threadIdx.x;
  const int half = (tid >> 4) & 1;    // K-half selector per WMMA lane layout
  const int l15  = tid & 15;
  const int wave = tid >> 5;
  const int wm   = wave & 3;          // 4 waves along M
  const int wn   = wave >> 2;         // 2 waves along N

  const int gm0 = blockIdx.y * BM;
  const int gn0 = blockIdx.x * BN;

  v8f zero = {};
  v8f m1[2][2], m2[2][2], m3[2][2];
#pragma unroll
  for (int mt = 0; mt < 2; ++mt)
#pragma unroll
    for (int nt = 0; nt < 2; ++nt) { m1[mt][nt] = zero; m2[mt][nt] = zero; m3[mt][nt] = zero; }

  // Issue all async B128 copies for K-stage k0 into LDS buffer `buf`.
  auto stage = [&](int k0, int buf) {
    float* As_r = lsA[buf][0];
    float* As_i = lsA[buf][1];
    float* Bs_r = lsB[buf][0];
    float* Bs_i = lsB[buf][1];
    if (!conjA) {
      // A tile rows gm0..+127, cols k0..+15 -> LDS [m][k]
      const int row = tid >> 1;
      const int cb  = (tid & 1) * 8;
      const size_t g = (size_t)(gm0 + row) * DIM + (size_t)(k0 + cb);
      async_b128(&As_r[row * APITCH + cb],     Ar + g);
      async_b128(&As_r[row * APITCH + cb + 4], Ar + g + 4);
      async_b128(&As_i[row * APITCH + cb],     Ai + g);
      async_b128(&As_i[row * APITCH + cb + 4], Ai + g + 4);
    } else {
      // A^H: stage raw rows k0..+15, cols gm0..+127 -> LDS [k][m]; transpose at frag read
      const int kr = tid >> 4;
      const int cb = (tid & 15) * 8;
      const size_t g = (size_t)(k0 + kr) * DIM + (size_t)(gm0 + cb);
      async_b128(&As_r[kr * TAPITCH + cb],     Ar + g);
      async_b128(&As_r[kr * TAPITCH + cb + 4], Ar + g + 4);
      async_b128(&As_i[kr * TAPITCH + cb],     Ai + g);
      async_b128(&As_i[kr * TAPITCH + cb + 4], Ai + g + 4);
    }
    {
      const int kr = tid >> 4;
      const int bc = (tid & 15) * 4;
      const size_t g = (size_t)(k0 + kr) * N + (size_t)(gn0 + bc);
      async_b128(&Bs_r[kr * BPITCH + bc], Br + g);
      async_b128(&Bs_i[kr * BPITCH + bc], Bi + g);
    }
  };

  stage(0, 0);
  wait_async0();
  __syncthreads();

  const int nstages = DIM / BK;
  int cur = 0;
  for (int s = 0; s < nstages; ++s) {
    if (s + 1 < nstages) stage((s + 1) * BK, cur ^ 1);  // overlap with WMMA below

    const float* As_r = lsA[cur][0];
    const float* As_i = lsA[cur][1];
    const float* Bs_r = lsB[cur][0];
    const float* Bs_i = lsB[cur][1];

#pragma unroll
    for (int kk = 0; kk < BK; kk += 4) {
      const int kidx = kk + half * 2;
      v2f ar[2], ai[2], as_[2], br[2], bi[2], bs[2];
#pragma unroll
      for (int mt = 0; mt < 2; ++mt) {
        const int row = wm * 32 + mt * 16 + l15;
        if (!conjA) {
          ar[mt] = *(const v2f*)&As_r[row * APITCH + kidx];
          ai[mt] = *(const v2f*)&As_i[row * APITCH + kidx];
        } else {
          v2f t;
          t.x = As_r[kidx * TAPITCH + row];
          t.y = As_r[(kidx + 1) * TAPITCH + row];
          ar[mt] = t;
          t.x = As_i[kidx * TAPITCH + row];
          t.y = As_i[(kidx + 1) * TAPITCH + row];
          ai[mt] = t;
        }
        as_[mt] = conjA ? (ar[mt] - ai[mt]) : (ar[mt] + ai[mt]);  // Ar + s*Ai
      }
#pragma unroll
      for (int nt = 0; nt < 2; ++nt) {
        const int col = wn * 32 + nt * 16 + l15;
        v2f t;
        t.x = Bs_r[kidx * BPITCH + col];
        t.y = Bs_r[(kidx + 1) * BPITCH + col];
        br[nt] = t;
        t.x = Bs_i[kidx * BPITCH + col];
        t.y = Bs_i[(kidx + 1) * BPITCH + col];
        bi[nt] = t;
        bs[nt] = br[nt] + bi[nt];
      }
#pragma unroll
      for (int mt = 0; mt < 2; ++mt)
#pragma unroll
        for (int nt = 0; nt < 2; ++nt) {
          m1[mt][nt] = wmma_f32x4(ar[mt],  br[nt], m1[mt][nt]);
          m2[mt][nt] = wmma_f32x4(ai[mt],  bi[nt], m2[mt][nt]);
          m3[mt][nt] = wmma_f32x4(as_[mt], bs[nt], m3[mt][nt]);
        }
    }

    wait_async0();     // our async writes to buf^1 have landed in LDS
    __syncthreads();   // ...and everyone is done reading buf
    cur ^= 1;
  }

  float a = alpha;
  if (alpha_ptr) a *= *alpha_ptr;
  const float sgn = conjA ? -1.0f : 1.0f;

#pragma unroll
  for (int mt = 0; mt < 2; ++mt)
#pragma unroll
    for (int nt = 0; nt < 2; ++nt) {
      const int gr0 = gm0 + wm * 32 + mt * 16 + half * 8;
      const int gc  = gn0 + wn * 32 + nt * 16 + l15;
      v8f c1 = m1[mt][nt], c2 = m2[mt][nt], c3 = m3[mt][nt];
#pragma unroll
      for (int i = 0; i < 8; ++i) {
        const int gr = gr0 + i;
        const float t2 = sgn * c2[i];
        float re = (c1[i] - t2) * a;
        float im = (c3[i] - c1[i] - t2) * a;
        if (diag_add != 0.0f && gr == gc) re += diag_add;
        Dr[(size_t)gr * N + gc] = re;
        Di[(size_t)gr * N + gc] = im;
      }
    }
}

// ---- deterministic two-pass Frobenius norm ----
__global__ __launch_bounds__(256)
void sumsq_kernel(const float* __restrict__ a, const float* __restrict__ b,
                  size_t n, float* __restrict__ partials)
{
  __shared__ float sm[256];
  size_t i = (size_t)blockIdx.x * blockDim.x + threadIdx.x;
  const size_t stride = (size_t)gridDim.x * blockDim.x;
  float s = 0.0f;
  for (; i < n; i += stride) { float x = a[i], y = b[i]; s += x * x + y * y; }
  sm[threadIdx.x] = s;
  __syncthreads();
  for (int o = 128; o > 0; o >>= 1) {
    if ((int)threadIdx.x < o) sm[threadIdx.x] += sm[threadIdx.x + o];
    __syncthreads();
  }
  if (threadIdx.x == 0) partials[blockIdx.x] = sm[0];
}

__global__ __launch_bounds__(512)
void invnorm_kernel(const float* __restrict__ partials, float* __restrict__ out)
{
  __shared__ float sm[512];
  const int t = threadIdx.x;
  sm[t] = partials[t];
  __syncthreads();
  for (int o = 256; o > 0; o >>= 1) {
    if (t < o) sm[t] += sm[t + o];
    __syncthreads();
  }
  if (t == 0) out[0] = 1.0f / sqrtf(sm[0]);
}

__global__ __launch_bounds__(256)
void scale2_kernel(const float* __restrict__ sr, const float* __restrict__ si,
                   float* __restrict__ dr, float* __restrict__ di,
                   const float* __restrict__ s, size_t n)
{
  const float sc = *s;
  size_t i = (size_t)blockIdx.x * blockDim.x + threadIdx.x;
  const size_t stride = (size_t)gridDim.x * blockDim.x;
  for (; i < n; i += stride) { dr[i] = sr[i] * sc; di[i] = si[i] * sc; }
}

__global__ __launch_bounds__(256)
void interleave_scale_kernel(const float* __restrict__ xr, const float* __restrict__ xi,
                             const float* __restrict__ s, float* __restrict__ out, size_t n)
{
  const float sc = *s;
  size_t i = (size_t)blockIdx.x * blockDim.x + threadIdx.x;
  const size_t stride = (size_t)gridDim.x * blockDim.x;
  for (; i < n; i += stride) {
    out[2 * i]     = xr[i] * sc;   // complex64 interleaved layout
    out[2 * i + 1] = xi[i] * sc;
  }
}

extern "C" void kernel_launch(void* const* d_in, const int* in_sizes, int n_in,
                              void* d_out, int out_size, void* d_ws, size_t ws_size,
                              hipStream_t stream)
{
  const size_t DD = (size_t)DIM * DIM;
  const size_t DB = (size_t)DIM * BATCH;
  float* w = (float*)d_ws;
  float* Yr = w; w += DD;  float* Yi = w; w += DD;
  float* Tr = w; w += DD;  float* Ti = w; w += DD;
  float* Zr = w; w += DD;  float* Zi = w; w += DD;
  float* XAr = w; w += DB; float* XAi = w; w += DB;
  float* XBr = w; w += DB; float* XBi = w; w += DB;
  float* partials = w; w += 512;
  float* sW = w; w += 1;
  float* sX = w; w += 1;

  const float* curXr = (const float*)d_in[0];
  const float* curXi = (const float*)d_in[1];
  const float* aptr = nullptr;   // deferred 1/||x|| scaling folded into next GEMM

  dim3 blk(256);

  for (int layer = 0; layer < 3; ++layer) {
    const float* Wr = (const float*)d_in[2 + 2 * layer];
    const float* Wi = (const float*)d_in[3 + 2 * layer];

    // Y0 = W / ||W||_F  (polar factor is scale-invariant; bounds sigma_max <= 1)
    sumsq_kernel<<<512, blk, 0, stream>>>(Wr, Wi, DD, partials);
    invnorm_kernel<<<1, 512, 0, stream>>>(partials, sW);
    scale2_kernel<<<1024, blk, 0, stream>>>(Wr, Wi, Yr, Yi, sW, DD);

    // Newton-Schulz polar iteration (GEMM-only unitarization):
    //   T = 3I - Y^H Y ;  Y <- 0.5 * Y * T
    float *yr = Yr, *yi = Yi, *zr = Zr, *zi = Zi;
    for (int it = 0; it < NS_ITERS; ++it) {
      cgemm_wmma_kernel<<<dim3(DIM / BN, DIM / BM), blk, 0, stream>>>(
          yr, yi, /*conjA=*/1, yr, yi, Tr, Ti, DIM, -1.0f, nullptr, 3.0f);
      cgemm_wmma_kernel<<<dim3(DIM / BN, DIM / BM), blk, 0, stream>>>(
          yr, yi, /*conjA=*/0, Tr, Ti, zr, zi, DIM, 0.5f, nullptr, 0.0f);
      float* t;
      t = yr; yr = zr; zr = t;
      t = yi; yi = zi; zi = t;
    }

    // X' = U * X, scaled by previous layer's 1/||x|| (linearity lets us defer it)
    float* nxtXr = (layer & 1) ? XBr : XAr;
    float* nxtXi = (layer & 1) ? XBi : XAi;
    cgemm_wmma_kernel<<<dim3(BATCH / BN, DIM / BM), blk, 0, stream>>>(
        yr, yi, /*conjA=*/0, curXr, curXi, nxtXr, nxtXi, BATCH, 1.0f, aptr, 0.0f);

    sumsq_kernel<<<512, blk, 0, stream>>>(nxtXr, nxtXi, DB, partials);
    invnorm_kernel<<<1, 512, 0, stream>>>(partials, sX);
    aptr = sX;
    curXr = nxtXr; curXi = nxtXi;
  }

  // Final normalization applied at the interleaving store.
  interleave_scale_kernel<<<2048, blk, 0, stream>>>(curXr, curXi, sX, (float*)d_out, DB);

  (void)in_sizes; (void)n_in; (void)out_size; (void)ws_size;
}